// SpanNER_16690242913141
// MI455X (gfx1250) — compile-verified
//
#include <hip/hip_runtime.h>

#define T_TOK  4096
#define H_DIM  1024
#define N_SPANS 65536
#define NLAB   9

typedef __attribute__((ext_vector_type(2))) float v2f;
typedef __attribute__((ext_vector_type(8))) float v8f;

// Workspace layout (d_ws):
//   Q : T_TOK x 32 f32  (cols 0-8 = tok.W1, 9-17 = tok.W2, 18-26 = tok.W3, 27-31 pad)
//   S3: (T_TOK+1) x 16 f32 (cols 0-8 = exclusive cumsum of Q cols 18-26)

// ---------------------------------------------------------------------------
// Kernel 1: Q = token_emb (4096x1024) @ Wview (1024x27, padded to 32)
// using V_WMMA_F32_16X16X4_F32. One wave per 16-row M tile, 2 N tiles.
// ---------------------------------------------------------------------------
__global__ __launch_bounds__(128) void spanner_gemm_wmma(
    const float* __restrict__ tok, const float* __restrict__ W,
    float* __restrict__ Q)
{
  const int wave = threadIdx.x >> 5;
  const int lane = threadIdx.x & 31;
  const int tile = blockIdx.x * 4 + wave;      // 0..255
  const int n    = lane & 15;                  // N column within tile / M row for A
  const int kh   = (lane >> 4) << 1;           // K sub-offset: 0 (lanes 0-15) or 2

  // A fragment source: one token row per lane (lanes 16-31 mirror rows 0-15)
  const float* arow = tok + (size_t)(tile * 16 + n) * H_DIM;

  // B: column j of the logical (1024 x 27) weight view:
  //    Bview[k][j] = W[((j/9)*1024 + k)*9 + (j%9)]
  const int j0 = n;            // N-tile 0: cols 0..15
  int       j1 = 16 + n;       // N-tile 1: cols 16..31 (>=27 zero-padded)
  const float sc1 = (j1 < 27) ? 1.0f : 0.0f;
  if (j1 > 26) j1 = 26;        // clamp so padded lanes still load in-bounds
  const int b0 = (j0 / 9) * (H_DIM * 9) + (j0 % 9);
  const int b1 = (j1 / 9) * (H_DIM * 9) + (j1 % 9);

  v8f acc0 = {}; v8f acc1 = {};
#pragma unroll 4
  for (int k = 0; k < H_DIM; k += 4) {
    v2f A;                                    // lanes<16: K=k,k+1; lanes>=16: K=k+2,k+3
    A.x = arow[k + kh + 0];
    A.y = arow[k + kh + 1];
    v2f B0, B1;                               // B row kh / kh+1 (mirrored across halves)
    B0.x = W[b0 + (k + kh + 0) * 9];
    B0.y = W[b0 + (k + kh + 1) * 9];
    B1.x = W[b1 + (k + kh + 0) * 9] * sc1;
    B1.y = W[b1 + (k + kh + 1) * 9] * sc1;
    acc0 = __builtin_amdgcn_wmma_f32_16x16x4_f32(false, A, false, B0,
                                                 (short)0, acc0, false, false);
    acc1 = __builtin_amdgcn_wmma_f32_16x16x4_f32(false, A, false, B1,
                                                 (short)0, acc1, false, false);
  }

  // C/D layout: VGPR v -> (M = v + 8*(lane>=16), N = lane&15)
  const int mbase = tile * 16 + ((lane >> 4) << 3);
#pragma unroll
  for (int v = 0; v < 8; ++v) {
    Q[(mbase + v) * 32 + n]      = acc0[v];
    Q[(mbase + v) * 32 + 16 + n] = acc1[v];
  }
}

// ---------------------------------------------------------------------------
// Kernel 2: S3 = exclusive prefix sum over T of Q[:, 18..26].
// Two-level scan: 32 chunks x 128 rows x 9 cols, one 288-thread workgroup.
// ---------------------------------------------------------------------------
__global__ __launch_bounds__(288) void spanner_scan(
    const float* __restrict__ Q, float* __restrict__ S3)
{
  __shared__ float part[288];
  const int tid  = threadIdx.x;
  const int c    = tid / 9;         // chunk 0..31
  const int l    = tid - c * 9;     // label col 0..8
  const int ROWS = T_TOK / 32;      // 128

  float s = 0.0f;
  for (int t = 0; t < ROWS; ++t)
    s += Q[(c * ROWS + t) * 32 + 18 + l];
  part[c * 9 + l] = s;
  __syncthreads();

  if (tid < 9) {                    // serial exclusive scan over 32 chunk partials
    float run = 0.0f;
    for (int cc = 0; cc < 32; ++cc) {
      float v = part[cc * 9 + tid];
      part[cc * 9 + tid] = run;
      run += v;
    }
  }
  __syncthreads();

  s = part[c * 9 + l];
  if (c == 0) S3[l] = 0.0f;         // S3 row 0 = zeros
  for (int t = 0; t < ROWS; ++t) {
    s += Q[(c * ROWS + t) * 32 + 18 + l];
    S3[(c * ROWS + t + 1) * 16 + l] = s;
  }
}

// ---------------------------------------------------------------------------
// Kernel 3: per-(span,label) gather from L2-resident tables + bias.
// ---------------------------------------------------------------------------
__global__ __launch_bounds__(256) void spanner_out(
    const int* __restrict__ spans, const float* __restrict__ Q,
    const float* __restrict__ S3, const float* __restrict__ bias,
    float* __restrict__ out)
{
  const int i = blockIdx.x * 256 + threadIdx.x;
  if (i >= N_SPANS * NLAB) return;
  const int n = i / 9;
  const int l = i - n * 9;
  const int s = spans[2 * n + 0];
  const int e = spans[2 * n + 1];
  const float len = (float)(e - s);
  const float v = Q[s * 32 + l]                        // h_start . W1
                + Q[(e - 1) * 32 + 9 + l]              // h_end . W2
                + (S3[e * 16 + l] - S3[s * 16 + l]) / len  // h_mean . W3
                + bias[l];
  out[i] = v;
}

// ---------------------------------------------------------------------------
extern "C" void kernel_launch(void* const* d_in, const int* in_sizes, int n_in,
                              void* d_out, int out_size, void* d_ws, size_t ws_size,
                              hipStream_t stream) {
  const float* tok   = (const float*)d_in[0];   // (4096, 1024) f32
  const int*   spans = (const int*)  d_in[1];   // (65536, 2) i32
  const float* W     = (const float*)d_in[2];   // (3072, 9) f32
  const float* bias  = (const float*)d_in[3];   // (9,) f32
  float* out = (float*)d_out;                   // (65536, 9) f32

  float* Q  = (float*)d_ws;                                            // 512 KB
  float* S3 = (float*)((char*)d_ws + (size_t)T_TOK * 32 * sizeof(float)); // 256 KB

  spanner_gemm_wmma<<<T_TOK / 64, 128, 0, stream>>>(tok, W, Q);
  spanner_scan<<<1, 288, 0, stream>>>(Q, S3);
  spanner_out<<<(N_SPANS * NLAB + 255) / 256, 256, 0, stream>>>(spans, Q, S3, bias, out);
}